// MKMIL_45947560133244
// MI455X (gfx1250) — compile-verified
//
#include <hip/hip_runtime.h>
#include <math.h>

typedef __attribute__((ext_vector_type(2))) float v2f;
typedef __attribute__((ext_vector_type(8))) float v8f;

#define L_TOK 8192

__device__ __forceinline__ float silu_f(float x) {
  return x / (1.f + __expf(-x));
}

__device__ __forceinline__ float act_apply(float v, int act) {
  if (act == 1) return fmaxf(v, 0.f);                       // relu
  if (act == 2) return tanhf(v);                            // tanh
  if (act == 3) return (v > 20.f) ? v : log1pf(__expf(v));  // softplus
  return v;
}

// ---------------------------------------------------------------------------
// WMMA GEMM: C[M,N] = act(A[M,K] @ W[N,K]^T + bias) + res
// One wave computes a 16 x (16*NT) slab of C with V_WMMA_F32_16X16X4_F32,
// reusing one A fragment across NT WMMAs. NT is a template parameter and the
// launch geometry guarantees every wave is fully in-bounds: no branches in
// the K-loop, EXEC is all-ones everywhere (WMMA requirement).
// A frag (16x4 f32): lanes 0-15 -> K={k0,k0+1}, lanes 16-31 -> K={k0+2,k0+3}
// B frag (4x16 f32): same half-wave K split, N = lane%16
// C/D: VGPR r -> row r (lanes 0-15) / row r+8 (lanes 16-31), N = lane%16
// ---------------------------------------------------------------------------
template <int NT>
__global__ void __launch_bounds__(128)
gemm_wmma_f32(const float* __restrict__ A, int lda,
              const float* __restrict__ W,
              const float* __restrict__ bias,
              const float* __restrict__ res,
              float* __restrict__ C,
              int N, int K, int act) {
  const int lane = threadIdx.x;          // 0..31
  const int m    = lane & 15;
  const int half = lane >> 4;            // 0 or 1
  const int koff = half << 1;            // 0 or 2
  const int colBase = (blockIdx.y * blockDim.y + threadIdx.y) * (16 * NT);
  const int row0 = blockIdx.x * 16;

  const float* Ap = A + (size_t)(row0 + m) * lda;
  const float* Wp[NT];
  #pragma unroll
  for (int t = 0; t < NT; ++t)
    Wp[t] = W + (size_t)(colBase + 16 * t + m) * K;

  v8f acc[NT];
  #pragma unroll
  for (int t = 0; t < NT; ++t)
    acc[t] = (v8f){0.f, 0.f, 0.f, 0.f, 0.f, 0.f, 0.f, 0.f};

  #pragma unroll 2
  for (int k0 = 0; k0 < K; k0 += 4) {
    const v2f a = *(const v2f*)(Ap + k0 + koff);
    v2f b[NT];
    #pragma unroll
    for (int t = 0; t < NT; ++t)
      b[t] = *(const v2f*)(Wp[t] + k0 + koff);
    #pragma unroll
    for (int t = 0; t < NT; ++t)
      acc[t] = __builtin_amdgcn_wmma_f32_16x16x4_f32(
          false, a, false, b[t], (short)0, acc[t], false, false);
  }

  #pragma unroll
  for (int t = 0; t < NT; ++t) {
    const int cc = colBase + 16 * t + m;
    const float bv = bias ? bias[cc] : 0.f;
    #pragma unroll
    for (int r = 0; r < 8; ++r) {
      const int rr = row0 + half * 8 + r;
      const size_t idx = (size_t)rr * N + cc;
      float v = act_apply(acc[t][r] + bv, act);
      if (res) v += res[idx];
      C[idx] = v;
    }
  }
}

// ---------------------------------------------------------------------------
// LayerNorm over last dim D (256 or 512), one row per 256-thread block
// ---------------------------------------------------------------------------
__global__ void __launch_bounds__(256)
layernorm_kernel(const float* __restrict__ x, const float* __restrict__ w,
                 const float* __restrict__ b, float* __restrict__ y, int D) {
  __shared__ float red[256];
  const int row = blockIdx.x;
  const float* xr = x + (size_t)row * D;

  float s = 0.f;
  for (int i = threadIdx.x; i < D; i += 256) s += xr[i];
  red[threadIdx.x] = s; __syncthreads();
  for (int off = 128; off > 0; off >>= 1) {
    if (threadIdx.x < off) red[threadIdx.x] += red[threadIdx.x + off];
    __syncthreads();
  }
  const float mu = red[0] / (float)D;
  __syncthreads();

  float v = 0.f;
  for (int i = threadIdx.x; i < D; i += 256) {
    float d = xr[i] - mu; v += d * d;
  }
  red[threadIdx.x] = v; __syncthreads();
  for (int off = 128; off > 0; off >>= 1) {
    if (threadIdx.x < off) red[threadIdx.x] += red[threadIdx.x + off];
    __syncthreads();
  }
  const float rstd = rsqrtf(red[0] / (float)D + 1e-5f);

  for (int i = threadIdx.x; i < D; i += 256)
    y[(size_t)row * D + i] = (xr[i] - mu) * rstd * w[i] + b[i];
}

// ---------------------------------------------------------------------------
// Depthwise causal conv1d (k=4) + bias + SiLU.
// xi = xz[:, 0:512] (stride 1024). xc[t,d] = silu(sum_j xi[t-3+j,d]*cw[d,j]+cb[d])
// ---------------------------------------------------------------------------
__global__ void __launch_bounds__(256)
conv_silu_kernel(const float* __restrict__ xz, const float* __restrict__ cw,
                 const float* __restrict__ cb, float* __restrict__ xc) {
  const int idx = blockIdx.x * 256 + threadIdx.x;
  if (idx >= L_TOK * 512) return;
  const int d = idx & 511;
  const int t = idx >> 9;
  float acc = cb[d];
  #pragma unroll
  for (int j = 0; j < 4; ++j) {
    const int tt = t - 3 + j;
    if (tt >= 0) acc += xz[(size_t)tt * 1024 + d] * cw[d * 4 + j];
  }
  xc[idx] = silu_f(acc);
}

// ---------------------------------------------------------------------------
// Selective scan: 8192 threads, each owns one (d, s) recurrence over L steps.
// Lanes [16*i .. 16*i+15] share one d; reduce over s with 16-wide shfl_xor.
// dt is post-softplus; dbl = [dt_raw(16) | B(16) | C(16)] stride 48.
// ---------------------------------------------------------------------------
__global__ void __launch_bounds__(256)
scan_kernel(const float* __restrict__ dt, const float* __restrict__ dbl,
            const float* __restrict__ xc, const float* __restrict__ A_log,
            float* __restrict__ y) {
  const int gid = blockIdx.x * 256 + threadIdx.x;  // 0..8191
  const int s = gid & 15;
  const int d = gid >> 4;
  const float Acoef = -__expf(A_log[d * 16 + s]);
  float h = 0.f;
  for (int t = 0; t < L_TOK; ++t) {
    const float dtv = dt[(size_t)t * 512 + d];
    const float Bv  = dbl[(size_t)t * 48 + 16 + s];
    const float Cv  = dbl[(size_t)t * 48 + 32 + s];
    const float xv  = xc[(size_t)t * 512 + d];
    const float dA  = __expf(dtv * Acoef);
    h = dA * h + dtv * Bv * xv;
    float contrib = h * Cv;
    #pragma unroll
    for (int off = 8; off > 0; off >>= 1)
      contrib += __shfl_xor(contrib, off, 16);
    if (s == 0) y[(size_t)t * 512 + d] = contrib;
  }
}

// y[t,d] = (y[t,d] + xc[t,d]*D[d]) * silu(z[t,d]),  z = xz[:, 512:1024]
__global__ void __launch_bounds__(256)
gate_kernel(float* __restrict__ y, const float* __restrict__ xc,
            const float* __restrict__ xz, const float* __restrict__ Dp) {
  const int idx = blockIdx.x * 256 + threadIdx.x;
  if (idx >= L_TOK * 512) return;
  const int d = idx & 511;
  const int t = idx >> 9;
  const float z = xz[(size_t)t * 1024 + 512 + d];
  y[idx] = (y[idx] + xc[idx] * Dp[d]) * silu_f(z);
}

// a[t] = dot(T[t, 0:128], w2) + b2 ; written to ws copy + d_out (A1 output)
__global__ void __launch_bounds__(128)
att2_kernel(const float* __restrict__ T, const float* __restrict__ w2,
            const float* __restrict__ b2, float* __restrict__ a_ws,
            float* __restrict__ a_out) {
  __shared__ float red[128];
  const int t = blockIdx.x;
  red[threadIdx.x] = T[(size_t)t * 128 + threadIdx.x] * w2[threadIdx.x];
  __syncthreads();
  for (int off = 64; off > 0; off >>= 1) {
    if (threadIdx.x < off) red[threadIdx.x] += red[threadIdx.x + off];
    __syncthreads();
  }
  if (threadIdx.x == 0) {
    const float r = red[0] + b2[0];
    a_ws[t] = r;
    a_out[t] = r;
  }
}

// softmax over L_TOK values (single block)
__global__ void __launch_bounds__(1024)
softmax_kernel(const float* __restrict__ a, float* __restrict__ aw) {
  __shared__ float red[1024];
  float m = -1e30f;
  for (int i = threadIdx.x; i < L_TOK; i += 1024) m = fmaxf(m, a[i]);
  red[threadIdx.x] = m; __syncthreads();
  for (int off = 512; off > 0; off >>= 1) {
    if (threadIdx.x < off)
      red[threadIdx.x] = fmaxf(red[threadIdx.x], red[threadIdx.x + off]);
    __syncthreads();
  }
  m = red[0]; __syncthreads();
  float s = 0.f;
  for (int i = threadIdx.x; i < L_TOK; i += 1024) {
    const float e = __expf(a[i] - m);
    aw[i] = e; s += e;
  }
  red[threadIdx.x] = s; __syncthreads();
  for (int off = 512; off > 0; off >>= 1) {
    if (threadIdx.x < off) red[threadIdx.x] += red[threadIdx.x + off];
    __syncthreads();
  }
  const float inv = 1.f / red[0];
  for (int i = threadIdx.x; i < L_TOK; i += 1024) aw[i] *= inv;
}

// feat[dcol] = sum_t aw[t] * H[t, dcol]   (H is 8192x512)
__global__ void __launch_bounds__(256)
feat_kernel(const float* __restrict__ aw, const float* __restrict__ H,
            float* __restrict__ feat) {
  __shared__ float red[256];
  const int dcol = blockIdx.x;
  float s = 0.f;
  for (int t = threadIdx.x; t < L_TOK; t += 256)
    s += aw[t] * H[(size_t)t * 512 + dcol];
  red[threadIdx.x] = s; __syncthreads();
  for (int off = 128; off > 0; off >>= 1) {
    if (threadIdx.x < off) red[threadIdx.x] += red[threadIdx.x + off];
    __syncthreads();
  }
  if (threadIdx.x == 0) feat[dcol] = red[0];
}

// logits, softmax, argmax (tiny)
__global__ void head_kernel(const float* __restrict__ feat,
                            const float* __restrict__ cw,
                            const float* __restrict__ cb,
                            float* __restrict__ out) {
  float l0 = cb[0], l1 = cb[1];
  for (int d = 0; d < 512; ++d) {
    l0 += feat[d] * cw[d];
    l1 += feat[d] * cw[512 + d];
  }
  out[0] = l0; out[1] = l1;
  const float m = fmaxf(l0, l1);
  const float e0 = __expf(l0 - m), e1 = __expf(l1 - m);
  const float s = e0 + e1;
  out[2] = e0 / s; out[3] = e1 / s;
  out[4] = (l1 > l0) ? 1.f : 0.f;   // argmax (first index on tie)
}

// ---------------------------------------------------------------------------
extern "C" void kernel_launch(void* const* d_in, const int* in_sizes, int n_in,
                              void* d_out, int out_size, void* d_ws, size_t ws_size,
                              hipStream_t stream) {
  (void)in_sizes; (void)n_in; (void)out_size; (void)ws_size;
  const float* data   = (const float*)d_in[0];
  const float* fc1_w1 = (const float*)d_in[1];
  const float* fc1_b1 = (const float*)d_in[2];
  const float* fc1_w2 = (const float*)d_in[3];
  const float* fc1_b2 = (const float*)d_in[4];
  const float* fc2_w  = (const float*)d_in[27];
  const float* fc2_b  = (const float*)d_in[28];
  const float* norm_w = (const float*)d_in[29];
  const float* norm_b = (const float*)d_in[30];
  const float* att_w1 = (const float*)d_in[31];
  const float* att_b1 = (const float*)d_in[32];
  const float* att_w2 = (const float*)d_in[33];
  const float* att_b2 = (const float*)d_in[34];
  const float* cls_w  = (const float*)d_in[35];
  const float* cls_b  = (const float*)d_in[36];

  float* out = (float*)d_out;
  // workspace carve-up (floats)
  float* R0  = (float*)d_ws;               // 8192*1024  xz
  float* R1  = R0 + (size_t)L_TOK * 1024;  // 8192*512   fc1 h1 / dt / HF
  float* R2  = R1 + (size_t)L_TOK * 512;   // 8192*512   xc / att_t
  float* R3  = R2 + (size_t)L_TOK * 512;   // 8192*512   y / fc2 out
  float* R4  = R3 + (size_t)L_TOK * 512;   // 8192*256   residual h
  float* R5  = R4 + (size_t)L_TOK * 256;   // 8192*256   hn
  float* R6  = R5 + (size_t)L_TOK * 256;   // 8192*64    dbl (48 used)
  float* R7a = R6 + (size_t)L_TOK * 64;    // 8192       att scores
  float* R7b = R7a + L_TOK;                // 8192       softmax weights

  float* out_logits = out;              // [0..4]: logits(2), prob(2), yhat(1)
  float* out_A1     = out + 5;          // 8192
  float* out_feat   = out + 5 + L_TOK;  // 512

  const int MT = L_TOK / 16;                 // 512 row tiles
  const int EW = (L_TOK * 512 + 255) / 256;  // elementwise grid
  // launch geometries (every wave fully in-bounds):
  const dim3 blk4(32, 4);   // NT=4: 256 cols per block
  const dim3 blk2(32, 2);   // NT=4: 128 cols per block
  const dim3 blk1(32, 1);   // NT=3: 48 cols per block

  // fc1: (8192,1024)@(512,1024)^T -> R1 ; then @(256,512)^T relu -> R4
  gemm_wmma_f32<4><<<dim3(MT, 2), blk4, 0, stream>>>(
      data, 1024, fc1_w1, fc1_b1, nullptr, R1, 512, 1024, 0);
  gemm_wmma_f32<4><<<dim3(MT, 1), blk4, 0, stream>>>(
      R1, 512, fc1_w2, fc1_b2, nullptr, R4, 256, 512, 1);

  for (int l = 0; l < 2; ++l) {
    const int p = 5 + 11 * l;
    const float* ln_w     = (const float*)d_in[p + 0];
    const float* ln_b     = (const float*)d_in[p + 1];
    const float* in_proj  = (const float*)d_in[p + 2];
    const float* conv_w   = (const float*)d_in[p + 3];
    const float* conv_b   = (const float*)d_in[p + 4];
    const float* x_proj   = (const float*)d_in[p + 5];
    const float* dt_w     = (const float*)d_in[p + 6];
    const float* dt_b     = (const float*)d_in[p + 7];
    const float* A_log    = (const float*)d_in[p + 8];
    const float* Dp       = (const float*)d_in[p + 9];
    const float* out_proj = (const float*)d_in[p + 10];

    layernorm_kernel<<<L_TOK, 256, 0, stream>>>(R4, ln_w, ln_b, R5, 256);
    // in_proj: (8192,256)@(1024,256)^T -> xz (R0), N=1024
    gemm_wmma_f32<4><<<dim3(MT, 4), blk4, 0, stream>>>(
        R5, 256, in_proj, nullptr, nullptr, R0, 1024, 256, 0);
    conv_silu_kernel<<<EW, 256, 0, stream>>>(R0, conv_w, conv_b, R2);
    // x_proj: (8192,512)@(48,512)^T -> dbl (R6, ldc=48), N=48 -> NT=3, 1 wave
    gemm_wmma_f32<3><<<dim3(MT, 1), blk1, 0, stream>>>(
        R2, 512, x_proj, nullptr, nullptr, R6, 48, 512, 0);
    // dt: softplus(dbl[:, :16]@(512,16)^T + dt_b) -> R1, N=512
    gemm_wmma_f32<4><<<dim3(MT, 2), blk4, 0, stream>>>(
        R6, 48, dt_w, dt_b, nullptr, R1, 512, 16, 3);
    // 8192 recurrences, 256 threads/block -> exactly 32 blocks
    scan_kernel<<<32, 256, 0, stream>>>(R1, R6, R2, A_log, R3);
    gate_kernel<<<EW, 256, 0, stream>>>(R3, R2, R0, Dp);
    // out_proj + residual -> R4, N=256
    gemm_wmma_f32<4><<<dim3(MT, 1), blk4, 0, stream>>>(
        R3, 512, out_proj, nullptr, R4, R4, 256, 512, 0);
  }

  // fc2 relu -> R3 ; layernorm -> R1 (HF)
  gemm_wmma_f32<4><<<dim3(MT, 2), blk4, 0, stream>>>(
      R4, 256, fc2_w, fc2_b, nullptr, R3, 512, 256, 1);
  layernorm_kernel<<<L_TOK, 256, 0, stream>>>(R3, norm_w, norm_b, R1, 512);
  // attention scores: N=128 -> NT=4 with 2 waves per block
  gemm_wmma_f32<4><<<dim3(MT, 1), blk2, 0, stream>>>(
      R1, 512, att_w1, att_b1, nullptr, R2, 128, 512, 2);
  att2_kernel<<<L_TOK, 128, 0, stream>>>(R2, att_w2, att_b2, R7a, out_A1);
  softmax_kernel<<<1, 1024, 0, stream>>>(R7a, R7b);
  feat_kernel<<<512, 256, 0, stream>>>(R7b, R1, out_feat);
  head_kernel<<<1, 1, 0, stream>>>(out_feat, cls_w, cls_b, out_logits);
}